// SupervisedClusterLoss2_3461743640879
// MI455X (gfx1250) — compile-verified
//
#include <hip/hip_runtime.h>
#include <math.h>

// ---------------------------------------------------------------------------
// SupervisedClusterLoss2 for MI455X (gfx1250).
//
// Cost model: streaming 512MB of pred @ 23.3 TB/s ~= 22us dominates; all the
// per-class math (100x64) and the 100x100 gram are noise. Kernel 1 is a pure
// bandwidth kernel (LDS-atomic segmented reduction, NT loads, global f32
// atomic flush). Kernel 2 finalizes with V_WMMA_F32_16X16X4_F32 for
// gram = mean @ mean.T on a 112-row zero-padded tile.
// ---------------------------------------------------------------------------

#define C_CLASSES 100
#define C_PAD     112          // 7 tiles of 16
#define NFEAT     64

typedef __attribute__((ext_vector_type(2))) float v2f;
typedef __attribute__((ext_vector_type(8))) float v8f;

// ws layout (floats):
//   [0,128)              counts (padded)
//   [128, 128+6400)      per-class sums
//   [6528, 12928)        per-class sum-of-squares
#define WS_CNT_OFF 0
#define WS_SUM_OFF 128
#define WS_SQ_OFF  (128 + C_CLASSES * NFEAT)
#define WS_FLOATS  (128 + 2 * C_CLASSES * NFEAT)

__global__ void zero_ws_kernel(float* __restrict__ ws, int n) {
    int i = blockIdx.x * blockDim.x + threadIdx.x;
    if (i < n) ws[i] = 0.0f;
}

// ---------------------------------------------------------------------------
// Kernel 1: segmented count/sum/sumsq. 512 threads = 8 rows x 64 features per
// iteration; fully coalesced 2KB tile loads, LDS fp32 atomics for the
// class-keyed scatter, single global-atomic flush per block.
// ---------------------------------------------------------------------------
__global__ __launch_bounds__(512) void seg_accum_kernel(
        const float* __restrict__ pred, const int* __restrict__ gidx,
        float* __restrict__ ws, int nrows) {
    __shared__ float lsum[C_CLASSES * NFEAT];   // 25600 B
    __shared__ float lsq[C_CLASSES * NFEAT];    // 25600 B
    __shared__ float lcnt[C_CLASSES];

    const int tid = threadIdx.x;
    for (int i = tid; i < C_CLASSES * NFEAT; i += 512) { lsum[i] = 0.0f; lsq[i] = 0.0f; }
    if (tid < C_CLASSES) lcnt[tid] = 0.0f;
    __syncthreads();

    const int f  = tid & 63;     // feature
    const int ty = tid >> 6;     // row-in-tile 0..7
    const long stride = (long)gridDim.x * 8;

    for (long r = (long)blockIdx.x * 8 + ty; r < nrows; r += stride) {
        const int c = gidx[r];                         // broadcast within wave
        const float v = __builtin_nontemporal_load(&pred[(size_t)r * NFEAT + f]);
        atomicAdd(&lsum[c * NFEAT + f], v);            // ds_add_f32
        atomicAdd(&lsq[c * NFEAT + f], v * v);
        if (f == 0) atomicAdd(&lcnt[c], 1.0f);
    }
    __syncthreads();

    float* gcnt = ws + WS_CNT_OFF;
    float* gsum = ws + WS_SUM_OFF;
    float* gsq  = ws + WS_SQ_OFF;
    for (int i = tid; i < C_CLASSES * NFEAT; i += 512) {
        unsafeAtomicAdd(&gsum[i], lsum[i]);            // global_atomic_add_f32
        unsafeAtomicAdd(&gsq[i],  lsq[i]);
    }
    if (tid < C_CLASSES) unsafeAtomicAdd(&gcnt[tid], lcnt[tid]);
}

// ---------------------------------------------------------------------------
// Kernel 2: finalize. mean/var/std/cov with VALU, gram = mean@mean.T with
// V_WMMA_F32_16X16X4_F32 over 28 lower-triangular 16x16 tiles (8 waves).
// ---------------------------------------------------------------------------
__global__ __launch_bounds__(256) void finalize_kernel(
        const float* __restrict__ ws, float* __restrict__ out) {
    __shared__ float smean[C_PAD * NFEAT];   // 28672 B, rows >=100 zeroed
    __shared__ float snorm[C_PAD];
    __shared__ float scolsq[NFEAT];
    __shared__ float rsum[256];
    __shared__ float rmin[256];

    const int tid = threadIdx.x;
    const float* gcnt = ws + WS_CNT_OFF;
    const float* gsum = ws + WS_SUM_OFF;
    const float* gsq  = ws + WS_SQ_OFF;

    // zero padding rows (classes 100..111)
    for (int i = tid; i < (C_PAD - C_CLASSES) * NFEAT; i += 256)
        smean[C_CLASSES * NFEAT + i] = 0.0f;

    // per-feature column: mean into LDS, std column-sum squared for cov
    if (tid < NFEAT) {
        const int ff = tid;
        float colsum = 0.0f;
        for (int c = 0; c < C_CLASSES; ++c) {
            const float cnt = gcnt[c];
            const float m   = gsum[c * NFEAT + ff] / cnt;
            const float var = (gsq[c * NFEAT + ff] - cnt * m * m) / (cnt - 1.0f);
            colsum += sqrtf(fmaxf(var, 0.0f));
            smean[c * NFEAT + ff] = m;
        }
        scolsq[ff] = colsum * colsum;
    }
    __syncthreads();

    // norms[c] = sum_f mean[c][f]^2  (diagonal of gram)
    if (tid < C_PAD) {
        float n = 0.0f;
        for (int ff = 0; ff < NFEAT; ++ff) { const float m = smean[tid * NFEAT + ff]; n += m * m; }
        snorm[tid] = n;
    }
    __syncthreads();

    // --- WMMA gram tiles -------------------------------------------------
    const int wave  = tid >> 5;
    const int lane  = tid & 31;
    const int lrow  = lane & 15;
    const int khalf = (lane >> 4) * 2;   // lanes 16..31 hold K=2,3 per ISA layout
    float sumDist = 0.0f;
    float minDist = INFINITY;

    for (int t = wave; t < 28; t += 8) {              // wave-uniform trip count
        // decode t -> (ci >= cj) over 7x7 lower triangle incl. diagonal
        int ci = 0, tt = t;
        while (tt >= ci + 1) { tt -= (ci + 1); ++ci; }
        const int cj = tt;

        const float* arow = &smean[(ci * 16 + lrow) * NFEAT + khalf];
        const float* brow = &smean[(cj * 16 + lrow) * NFEAT + khalf];
        v8f acc = {};
        for (int k0 = 0; k0 < NFEAT; k0 += 4) {
            v2f a; a.x = arow[k0]; a.y = arow[k0 + 1];
            v2f b; b.x = brow[k0]; b.y = brow[k0 + 1];
            acc = __builtin_amdgcn_wmma_f32_16x16x4_f32(
                false, a, false, b, (short)0, acc, false, false);
        }
        // D layout: VGPR r -> M = r (lanes 0-15) / r+8 (lanes 16-31), N = lane&15
        const int gn    = cj * 16 + lrow;
        const int mbase = ci * 16 + ((lane < 16) ? 0 : 8);
        for (int r = 0; r < 8; ++r) {
            const int gm = mbase + r;
            if (gm < C_CLASSES && gn < C_CLASSES && gn < gm) {
                const float d = snorm[gm] + snorm[gn] - 2.0f * acc[r];
                sumDist += d;
                minDist = fminf(minDist, d);
            }
        }
    }

    // block reduction
    rsum[tid] = sumDist;
    rmin[tid] = minDist;
    __syncthreads();
    for (int s = 128; s > 0; s >>= 1) {
        if (tid < s) {
            rsum[tid] += rsum[tid + s];
            rmin[tid] = fminf(rmin[tid], rmin[tid + s]);
        }
        __syncthreads();
    }

    if (tid == 0) {
        float cov = 0.0f;
        for (int ff = 0; ff < NFEAT; ++ff) cov += scolsq[ff];
        cov /= (float)C_CLASSES;
        const float npairs = (float)(C_CLASSES * (C_CLASSES - 1)) * 0.5f;
        const float mean_distance = rsum[0] / npairs;
        const float min_dist = rmin[0];
        const float dm = 8.0f - mean_distance;
        const float dn = 8.0f - min_dist;
        const float loss = 1.0f * cov + 0.1f * dm * dm + 0.0f * dn * dn;
        out[0] = loss;
        out[1] = mean_distance;
        out[2] = min_dist;
        out[3] = cov;
    }
}

extern "C" void kernel_launch(void* const* d_in, const int* in_sizes, int n_in,
                              void* d_out, int out_size, void* d_ws, size_t ws_size,
                              hipStream_t stream) {
    const float* pred = (const float*)d_in[0];
    const int*   gidx = (const int*)d_in[1];
    float* out = (float*)d_out;
    float* ws  = (float*)d_ws;
    const int nrows = in_sizes[1];   // 2,000,000

    zero_ws_kernel<<<(WS_FLOATS + 255) / 256, 256, 0, stream>>>(ws, WS_FLOATS);

    const int grid = 1024;           // grid-stride; ~244 row-tiles per block
    seg_accum_kernel<<<grid, 512, 0, stream>>>(pred, gidx, ws, nrows);

    finalize_kernel<<<1, 256, 0, stream>>>(ws, out);
}